// DenseNetCmaxGatedB2_33749853012203
// MI455X (gfx1250) — compile-verified
//
#include <hip/hip_runtime.h>
#include <hip/hip_bf16.h>

typedef __attribute__((ext_vector_type(2))) float v2f;
typedef __attribute__((ext_vector_type(8))) float v8f;

#define XST 72              // padded LDS row stride for the x tile (69 cols used)
#define NREG 1156           // 34*34 n0-region pixels (32x32 interior + 1 halo ring)
#define NGROUP 73           // ceil(1156/16) pixel groups per tile
#define NEG_INF (-3.402823466e38f)

// fast sigmoid: v_exp_f32 + v_rcp_f32 (avoid the IEEE divide expansion)
__device__ __forceinline__ float sigm(float v) {
    return __builtin_amdgcn_rcpf(1.0f + __expf(-v));
}

// CDNA5 async global->LDS copy (ASYNCcnt-tracked, bypasses VGPRs).
__device__ __forceinline__ void async_ld_b32(unsigned lds_byte, const float* sbase, int goff_bytes) {
    asm volatile("global_load_async_to_lds_b32 %0, %1, %2"
                 :: "v"(lds_byte), "v"(goff_bytes), "s"(sbase)
                 : "memory");
}

__global__ __launch_bounds__(256) void fused_transition(
    const float* __restrict__ x,        // [B,C,128,128]
    const float* __restrict__ maxgate,  // [C,1,3,3]
    const float* __restrict__ mb,       // [C]
    const float* __restrict__ pconvs,   // [C,1,3,3,4]
    const float* __restrict__ pbs,      // [C,4]
    const float* __restrict__ pgates,   // [C,1,3,3,3]
    const float* __restrict__ gbs,      // [C,3]
    float* __restrict__ out)            // [B,C,64,64]
{
    __shared__ float xs[69 * XST];      // x halo tile
    __shared__ float n0s[34 * 35];      // n0 with 1-px halo (zero = conv padding)
    __shared__ float n1s[32 * 32];
    __shared__ float omx[32 * 32];      // maxpool * maxgate-conv branch

    const int tid  = threadIdx.x;
    const int lane = tid & 31;
    const int wave = tid >> 5;
    const int bz   = blockIdx.y;        // b*C + c
    const int cch  = bz & 255;
    const int i0   = (blockIdx.x >> 1) * 32;
    const int j0   = (blockIdx.x & 1) * 32;

    const float* xc = x + (size_t)bz * (128 * 128);

    // ---- Phase 0: async-stream x tile rows [2*i0-3, 2*i0+65] x cols [2*j0-3, 2*j0+65] into LDS
    const int R0 = 2 * i0 - 3, C0 = 2 * j0 - 3;
    const unsigned xs_base = (unsigned)(unsigned long long)&xs[0];
    for (int e = tid; e < 69 * 69; e += 256) {
        int r = e / 69, c = e - r * 69;
        int gr = R0 + r, gc = C0 + c;
        int li = r * XST + c;
        if ((unsigned)gr < 128u && (unsigned)gc < 128u) {
            async_ld_b32(xs_base + (unsigned)li * 4u, xc, (gr * 128 + gc) * 4);
        } else {
            xs[li] = 0.0f;              // conv zero-padding; maxpool re-checks bounds itself
        }
    }
    asm volatile("s_wait_asynccnt 0" ::: "memory");

    // ---- uniform per-channel biases (scalar loads)
    const float mbv = mb[cch];
    const float pb0 = pbs[cch * 4 + 0], pb1 = pbs[cch * 4 + 1];
    const float pb2 = pbs[cch * 4 + 2], pb3 = pbs[cch * 4 + 3];
    const float gb0 = gbs[cch * 3 + 0], gb1 = gbs[cch * 3 + 1], gb2 = gbs[cch * 3 + 2];

    // ---- per-lane A-matrix (filters x taps) + LDS tap offsets for B fill
    // filter m: 0=maxgate, 1..4=pconv0..3, 5=pgate[...,0], 6=pgate[...,2]; rows 7..15 zero
    const int half = lane >> 4;
    const int m    = lane & 15;
    v2f aw0{}, aw1{}, aw2{};
    int bo[6];
#pragma unroll
    for (int jj = 0; jj < 3; ++jj) {
        float wv[2];
#pragma unroll
        for (int q = 0; q < 2; ++q) {
            int t = 4 * jj + 2 * half + q;
            float w = 0.0f;
            if (t < 9 && m < 7) {
                if (m == 0)      w = maxgate[cch * 9 + t];
                else if (m <= 4) w = pconvs[(cch * 9 + t) * 4 + (m - 1)];
                else if (m == 5) w = pgates[(cch * 9 + t) * 3 + 0];
                else             w = pgates[(cch * 9 + t) * 3 + 2];
            }
            wv[q] = w;
            int dy = t / 3;
            bo[jj * 2 + q] = (t < 9) ? (dy * XST + (t - dy * 3)) : -1;
        }
        if (jj == 0)      { aw0.x = wv[0]; aw0.y = wv[1]; }
        else if (jj == 1) { aw1.x = wv[0]; aw1.y = wv[1]; }
        else              { aw2.x = wv[0]; aw2.y = wv[1]; }
    }
    __syncthreads();

    // ---- Phase A: 7 stride-2 convs via 3x V_WMMA_F32_16X16X4_F32 per 16-pixel group
    for (int g = wave; g < NGROUP; g += 8) {
        int lin0 = g * 16 + m;                       // pixel id (N = lane&15)
        int lin  = lin0 < NREG ? lin0 : NREG - 1;
        int pi   = lin / 34;
        int pj   = lin - pi * 34;
        int base = (2 * pi) * XST + 2 * pj;          // top-left of 3x3 patch in LDS

        v2f b0, b1, b2;
        b0.x = bo[0] >= 0 ? xs[base + bo[0]] : 0.0f;
        b0.y = bo[1] >= 0 ? xs[base + bo[1]] : 0.0f;
        b1.x = bo[2] >= 0 ? xs[base + bo[2]] : 0.0f;
        b1.y = bo[3] >= 0 ? xs[base + bo[3]] : 0.0f;
        b2.x = bo[4] >= 0 ? xs[base + bo[4]] : 0.0f;
        b2.y = bo[5] >= 0 ? xs[base + bo[5]] : 0.0f;

        v8f acc = {};
        acc = __builtin_amdgcn_wmma_f32_16x16x4_f32(false, aw0, false, b0, (short)0, acc, false, false);
        acc = __builtin_amdgcn_wmma_f32_16x16x4_f32(false, aw1, false, b1, (short)0, acc, false, false);
        acc = __builtin_amdgcn_wmma_f32_16x16x4_f32(false, aw2, false, b2, (short)0, acc, false, false);

        // lanes 0..15 own pixel n=lane: acc[r] = conv result of filter r
        if (lane < 16 && lin0 < NREG) {
            int gi = i0 - 1 + pi;
            int gj = j0 - 1 + pj;
            float n0v = 0.0f;
            if ((unsigned)gi < 64u && (unsigned)gj < 64u) {
                float g0 = sigm(acc[5] + gb0);
                n0v = g0 * (acc[1] + pb0) + (1.0f - g0) * (acc[2] + pb1);
            }
            n0s[pi * 35 + pj] = n0v;
            if (pi >= 1 && pi <= 32 && pj >= 1 && pj <= 32) {
                float g1 = sigm(acc[6] + gb1);
                n1s[(pi - 1) * 32 + (pj - 1)] =
                    g1 * (acc[3] + pb2) + (1.0f - g1) * (acc[4] + pb3);
                float mx = NEG_INF;
#pragma unroll
                for (int dy = 0; dy < 3; ++dy)
#pragma unroll
                    for (int dx = 0; dx < 3; ++dx) {
                        int xr = 2 * gi - 1 + dy, xcl = 2 * gj - 1 + dx;
                        float v = ((unsigned)xr < 128u && (unsigned)xcl < 128u)
                                      ? xs[(2 * pi + dy) * XST + (2 * pj + dx)] : NEG_INF;
                        mx = fmaxf(mx, v);
                    }
                omx[(pi - 1) * 32 + (pj - 1)] = mx * (acc[0] + mbv);
            }
        }
    }
    __syncthreads();

    // ---- Phase B: stride-1 node gate conv on n0 (from LDS) + final combine
    float wn[9];
#pragma unroll
    for (int t = 0; t < 9; ++t) wn[t] = pgates[(cch * 9 + t) * 3 + 2];  // uniform -> SGPRs
    float* outp = out + (size_t)bz * (64 * 64);
    for (int e = tid; e < 1024; e += 256) {
        int ii = e >> 5, jc = e & 31;
        float a = gb2;
#pragma unroll
        for (int dy = 0; dy < 3; ++dy)
#pragma unroll
            for (int dx = 0; dx < 3; ++dx)
                a = fmaf(wn[dy * 3 + dx], n0s[(ii + dy) * 35 + (jc + dx)], a);
        float gg = sigm(a);
        float outv = omx[e] + n0s[(ii + 1) * 35 + (jc + 1)] * gg + n1s[e] * (1.0f - gg);
        outp[(i0 + ii) * 64 + (j0 + jc)] = outv;
    }
}

extern "C" void kernel_launch(void* const* d_in, const int* in_sizes, int n_in,
                              void* d_out, int out_size, void* d_ws, size_t ws_size,
                              hipStream_t stream) {
    const float* x       = (const float*)d_in[0];
    const float* maxgate = (const float*)d_in[1];
    const float* mb      = (const float*)d_in[2];
    const float* pconvs  = (const float*)d_in[3];
    const float* pbs     = (const float*)d_in[4];
    const float* pgates  = (const float*)d_in[5];
    const float* gbs     = (const float*)d_in[6];
    float* out = (float*)d_out;

    dim3 grid(4, 16 * 256);   // 2x2 spatial tiles of 32x32, one block per (b,c,tile)
    fused_transition<<<grid, 256, 0, stream>>>(x, maxgate, mb, pconvs, pbs, pgates, gbs, out);

    (void)in_sizes; (void)n_in; (void)out_size; (void)d_ws; (void)ws_size;
}